// GEMV_42047729828312
// MI455X (gfx1250) — compile-verified
//
#include <hip/hip_runtime.h>
#include <cstdint>

// y = alpha * (A @ x) + beta * b    A: 16384x16384 f32 row-major
//
// MI455X (gfx1250, wave32) memory-bound GEMV:
//  - A streamed via GLOBAL_LOAD_ASYNC_TO_LDS_B128 (ASYNCcnt), NT temporal hint
//  - double-buffered LDS tiles: async engine fetches stage s+1 while WMMA eats stage s
//  - WMMA f32 16x16x4 accumulates 16 row dot-products (x occupies B column 0)
//  - LDS row stride 36 floats -> bank-conflict-free WMMA-layout ds_load_b64
//  - deterministic: no atomics; cross-wave reduce via LDS

typedef __attribute__((ext_vector_type(2))) float v2f;
typedef __attribute__((ext_vector_type(8))) float v8f;

#define ROWS 16384
#define COLS 16384
#define WAVES 8                       // waves per workgroup
#define TILE_M 16                     // rows per workgroup (one WMMA M tile)
#define KC 32                         // columns staged per stage
#define LDS_STRIDE 36                 // KC + 4 pad (gcd(36,64)=4 -> conflict-free)
#define COLS_PER_WAVE (COLS / WAVES)  // 2048
#define STAGES (COLS_PER_WAVE / KC)   // 64

__global__ __launch_bounds__(256) void gemv_wmma_async_kernel(
    const float* __restrict__ alpha, const float* __restrict__ A,
    const float* __restrict__ x, const float* __restrict__ beta,
    const float* __restrict__ b, float* __restrict__ y)
{
  __shared__ float sA[WAVES][2][TILE_M * LDS_STRIDE];  // 36 KB
  __shared__ float sX[WAVES][2][KC];                   // 2 KB
  __shared__ float sPart[WAVES * TILE_M];              // 0.5 KB

  const int tid  = threadIdx.x;
  const int wave = tid >> 5;
  const int lane = tid & 31;
  const int half = lane >> 4;   // 0: lanes 0-15, 1: lanes 16-31
  const int l16  = lane & 15;

  const int rowBase = blockIdx.x * TILE_M;
  const float* __restrict__ Abase =
      A + (size_t)rowBase * COLS + wave * COLS_PER_WAVE;
  const float* __restrict__ xbase = x + wave * COLS_PER_WAVE;

  // per-lane piece of each async stage: 4 instructions x (4 rows, 16B/lane)
  const int arow = lane >> 3;         // 0..3 within each 4-row group
  const int acol = (lane & 7) * 4;    // 0,4,...,28 floats

  v8f acc = {};  // D/C: 16x16 f32, column 0 = our 16 dot products

  // ---- stage prefetch: A tile async -> LDS, x chunk via normal path ----
  auto prefetch = [&](int s, int buf) {
    const int k0 = s * KC;
#pragma unroll
    for (int i = 0; i < 4; ++i) {
      const int row = i * 4 + arow;
      const float* gptr = Abase + (size_t)row * COLS + k0 + acol;
      const unsigned lptr =
          (unsigned)(uintptr_t)&sA[wave][buf][row * LDS_STRIDE + acol];
      asm volatile("global_load_async_to_lds_b128 %0, %1, off th:TH_LOAD_NT"
                   :: "v"(lptr), "v"(gptr) : "memory");
    }
    sX[wave][buf][lane] = xbase[k0 + lane];  // 32 f32, cache-friendly
  };

  prefetch(0, 0);

  for (int s = 0; s < STAGES; ++s) {
    const int buf = s & 1;
    if (s + 1 < STAGES) {
      prefetch(s + 1, buf ^ 1);
      // 4 newly issued remain outstanding; in-order completion means the
      // previous stage's 4 transfers have landed in LDS.
      asm volatile("s_wait_asynccnt 0x4" ::: "memory");
    } else {
      asm volatile("s_wait_asynccnt 0x0" ::: "memory");
    }

    // ---- 8 WMMA steps across K ----
    // A operand layout (f32 16x4): lane l%16 = row, half-wave supplies
    // K-pair {0,1} (lanes 0-15) or {2,3} (lanes 16-31).
#pragma unroll
    for (int j = 0; j < KC / 4; ++j) {
      const int kk = j * 4 + half * 2;
      const v2f av = *(const v2f*)&sA[wave][buf][l16 * LDS_STRIDE + kk];
      const v2f xv = *(const v2f*)&sX[wave][buf][kk];  // broadcast per half
      v2f bv;
      bv.x = (l16 == 0) ? xv.x : 0.0f;  // B column 0 = x, other columns 0
      bv.y = (l16 == 0) ? xv.y : 0.0f;
      acc = __builtin_amdgcn_wmma_f32_16x16x4_f32(
          false, av, false, bv, (short)0, acc, false, false);
    }
  }

  // ---- D column 0: lane 0 holds M=0..7 in acc[0..7], lane 16 holds M=8..15
  if (l16 == 0) {
#pragma unroll
    for (int m = 0; m < 8; ++m)
      sPart[wave * TILE_M + half * 8 + m] = acc[m];
  }
  __syncthreads();

  // ---- cross-wave reduce + affine epilogue ----
  if (tid < TILE_M) {
    float sum = 0.0f;
#pragma unroll
    for (int w = 0; w < WAVES; ++w) sum += sPart[w * TILE_M + tid];
    y[rowBase + tid] = alpha[0] * sum + beta[0] * b[rowBase + tid];
  }
}

extern "C" void kernel_launch(void* const* d_in, const int* in_sizes, int n_in,
                              void* d_out, int out_size, void* d_ws, size_t ws_size,
                              hipStream_t stream) {
  (void)in_sizes; (void)n_in; (void)d_ws; (void)ws_size; (void)out_size;
  const float* alpha = (const float*)d_in[0];
  const float* A     = (const float*)d_in[1];
  const float* x     = (const float*)d_in[2];
  const float* beta  = (const float*)d_in[3];
  const float* b     = (const float*)d_in[4];
  float* y = (float*)d_out;

  dim3 grid(ROWS / TILE_M);   // 1024 workgroups
  dim3 block(32 * WAVES);     // 256 threads = 8 waves (wave32)
  gemv_wmma_async_kernel<<<grid, block, 0, stream>>>(alpha, A, x, beta, b, y);
}